// DEFT_60790967108354
// MI455X (gfx1250) — compile-verified
//
#include <hip/hip_runtime.h>

typedef __attribute__((ext_vector_type(2))) float v2f;
typedef __attribute__((ext_vector_type(4))) float v4f;
typedef __attribute__((ext_vector_type(8))) float v8f;

#define INDIM 128   // in_dim = H*D
#define HEADS 8
#define HDIM  16

// ---------------------------------------------------------------------------
// Fused projection GEMM: Y[m][n] = scale * sum_k X[m][k] * W[n][k]
// One 256-thread block (8 waves) -> 16 rows x 128 cols of Y.
// Wave w handles N-tile [w*16, w*16+16); K=128 swept in 32 steps of
// V_WMMA_F32_16X16X4_F32 (fp32 A/B, fp32 accum -> matches fp32 reference).
// ---------------------------------------------------------------------------
__global__ __launch_bounds__(256)
void gat_gemm_wmma(const float* __restrict__ X, const float* __restrict__ W,
                   float* __restrict__ Y, float scale, int nrows) {
  const int m0   = blockIdx.x * 16;
  const int wave = threadIdx.x >> 5;
  const int lane = threadIdx.x & 31;
  const int n0   = wave * 16;
  const int mr   = lane & 15;   // A row within tile / B column within tile
  const int g    = lane >> 4;   // lane-group: selects K sub-pair

  int mload = m0 + mr;
  if (mload >= nrows) mload = nrows - 1;          // clamp (N divisible by 16 anyway)
  const float* Xrow = X + (size_t)mload * INDIM;
  const float* Wrow = W + (size_t)(n0 + mr) * INDIM;

  v8f acc = {0.f, 0.f, 0.f, 0.f, 0.f, 0.f, 0.f, 0.f};
#pragma unroll
  for (int k0 = 0; k0 < INDIM; k0 += 4) {
    // ISA layout: VGPR0 = A[m][k0+2g], VGPR1 = A[m][k0+2g+1]  (same for B=W^T)
    v2f a = *(const v2f*)(Xrow + k0 + 2 * g);
    v2f b = *(const v2f*)(Wrow + k0 + 2 * g);
    acc = __builtin_amdgcn_wmma_f32_16x16x4_f32(
        /*neg_a=*/false, a, /*neg_b=*/false, b,
        /*c_mod=*/(short)0, acc, /*reuse_a=*/false, /*reuse_b=*/false);
  }

  // D element r sits at (row = m0 + 8*g + r, col = n0 + mr)
  float* Yout = Y + (size_t)(m0 + 8 * g) * INDIM + n0 + mr;
#pragma unroll
  for (int r = 0; r < 8; ++r) {
    if (m0 + 8 * g + r < nrows) Yout[(size_t)r * INDIM] = acc[r] * scale;
  }
}

// ---------------------------------------------------------------------------
// Edge phase: one wave per edge.
//   lane l covers elements [4l, 4l+4) of the 128-wide feature row
//   => head h = l>>2, dims d = (l&3)*4 .. +4
// Gathers are fully coalesced 512B (32 x float4). Head dot product reduced
// over the 4 lanes of each head group via shfl_xor. Kh was pre-scaled by
// 1/sqrt(D), so p is already the scaled score.
// ---------------------------------------------------------------------------
__global__ __launch_bounds__(256)
void gat_edge(const float* __restrict__ Qh, const float* __restrict__ Kh,
              const float* __restrict__ Vh, const int* __restrict__ esrc,
              const int* __restrict__ edst, float* __restrict__ outWV,
              float* __restrict__ z, int nedges) {
  const int e = blockIdx.x * 8 + (threadIdx.x >> 5);
  if (e >= nedges) return;
  const int lane = threadIdx.x & 31;
  const int src  = esrc[e];
  const int dst  = edst[e];

  const v4f q = *(const v4f*)(Qh + (size_t)dst * INDIM + lane * 4);
  const v4f k = *(const v4f*)(Kh + (size_t)src * INDIM + lane * 4);
  const v4f v = *(const v4f*)(Vh + (size_t)src * INDIM + lane * 4);

  float p = q.x * k.x + q.y * k.y + q.z * k.z + q.w * k.w;
  p += __shfl_xor(p, 1, 32);
  p += __shfl_xor(p, 2, 32);          // all 4 lanes of the head group hold s/sqrt(D)

  p = fminf(5.0f, fmaxf(-5.0f, p));
  const float a = __expf(p);

  float* o = outWV + (size_t)dst * INDIM + lane * 4;
  unsafeAtomicAdd(o + 0, a * v.x);
  unsafeAtomicAdd(o + 1, a * v.y);
  unsafeAtomicAdd(o + 2, a * v.z);
  unsafeAtomicAdd(o + 3, a * v.w);
  if ((lane & 3) == 0) {
    unsafeAtomicAdd(z + (size_t)dst * HEADS + (lane >> 2), a);
  }
}

// ---------------------------------------------------------------------------
// Normalize in place: out[n][h][d] /= z[n][h]
// ---------------------------------------------------------------------------
__global__ __launch_bounds__(256)
void gat_norm(float* __restrict__ out, const float* __restrict__ z, int total) {
  const int i = blockIdx.x * 256 + threadIdx.x;
  if (i >= total) return;
  const int n = i >> 7;          // /128
  const int h = (i >> 4) & 7;    // head
  out[i] = out[i] / z[n * HEADS + h];
}

extern "C" void kernel_launch(void* const* d_in, const int* in_sizes, int n_in,
                              void* d_out, int out_size, void* d_ws, size_t ws_size,
                              hipStream_t stream) {
  const float* query = (const float*)d_in[0];
  const float* keyx  = (const float*)d_in[1];
  const float* value = (const float*)d_in[2];
  const float* WQ    = (const float*)d_in[3];
  const float* WK    = (const float*)d_in[4];
  const float* WV    = (const float*)d_in[5];
  const int*   esrc  = (const int*)d_in[6];
  const int*   edst  = (const int*)d_in[7];

  const int N = in_sizes[0] / INDIM;   // 50000
  const int E = in_sizes[6];           // 800000
  float* out = (float*)d_out;

  // Workspace layout: Qh | Kh | Vh | z   (25.6MB x3 + 1.6MB = 78.4MB)
  float* Qh = (float*)d_ws;
  float* Kh = Qh + (size_t)N * INDIM;
  float* Vh = Kh + (size_t)N * INDIM;
  float* z  = Vh + (size_t)N * INDIM;

  // Accumulators must be zeroed every call (atomics accumulate into them).
  hipMemsetAsync(d_out, 0, (size_t)out_size * sizeof(float), stream);
  hipMemsetAsync(z, 0, (size_t)N * HEADS * sizeof(float), stream);

  const float inv_scale = 0.25f;  // 1/sqrt(D), folded into Kh at store time
  dim3 blk(256);
  dim3 gg((N + 15) / 16);
  gat_gemm_wmma<<<gg, blk, 0, stream>>>(query, WQ, Qh, 1.0f, N);
  gat_gemm_wmma<<<gg, blk, 0, stream>>>(keyx,  WK, Kh, inv_scale, N);
  gat_gemm_wmma<<<gg, blk, 0, stream>>>(value, WV, Vh, 1.0f, N);

  gat_edge<<<dim3((E + 7) / 8), blk, 0, stream>>>(Qh, Kh, Vh, esrc, edst, out, z, E);
  gat_norm<<<dim3((N * INDIM + 255) / 256), blk, 0, stream>>>(out, z, N * INDIM);
}